// AdaptiveConv2d_40132174414046
// MI455X (gfx1250) — compile-verified
//
#include <hip/hip_runtime.h>
#include <math.h>

typedef float v2f __attribute__((ext_vector_type(2)));
typedef float v8f __attribute__((ext_vector_type(8)));

#if defined(__has_builtin)
# if __has_builtin(__builtin_amdgcn_global_load_async_to_lds_b32)
#  define USE_ASYNC_LDS 1
# endif
# if __has_builtin(__builtin_amdgcn_wmma_f32_16x16x4_f32)
#  define USE_WMMA 1
# endif
#endif

// builtin prototype wants plain int* in the named address spaces
typedef int __attribute__((address_space(1)))* gas1_int_ptr;
typedef int __attribute__((address_space(3)))* as3_int_ptr;

namespace {
constexpr int kBatch = 4, kChan = 3, kH = 256, kW = 256;
constexpr int kPad  = 4;
constexpr int kTile = 16;
constexpr int kHalo = kTile + 2 * kPad;   // 24
constexpr int kNT   = kHalo * kHalo;      // 576 floats per channel
constexpr float kSigMin = 0.5f, kSigMax = 10.0f;
constexpr float kPiF = 3.14159265358979323846f;
}

// ring index for each r^2 value (r^2 in [0,32]); r^2==0 is the center tap
// (sentinel 15, never matched by non-center taps). 14 real rings: indices 0..13.
__constant__ int c_ridx[33] = {
  15, 0, 1, 0, 2, 3, 0, 0, 4, 5, 6, 0, 0, 7, 0, 0,
   8, 9,10, 0,11, 0, 0, 0, 0,12, 0, 0, 0, 0, 0, 0, 13 };
// ring index -> r^2 value (rows 14,15 are zero-padded rings; q=1 keeps g finite)
__constant__ float c_ringq[16] = {1,2,4,5,8,9,10,13,16,17,18,20,25,32,1,1};

__device__ __forceinline__ int refl(int i, int n) {
  i = (i < 0) ? -i : i;                    // jnp.pad 'reflect', pad (4) < n
  return (i >= n) ? (2 * n - 2 - i) : i;
}

__global__ __launch_bounds__(256) void adaptive_log_conv(
    const float* __restrict__ x, const float* __restrict__ foa,
    float* __restrict__ out)
{
  __shared__ float tile[kChan * kNT];      // 24x24x3 halo tile, 6912 B

  const int bx = blockIdx.x, by = blockIdx.y, b = blockIdx.z;
  const int x0 = bx * kTile, y0 = by * kTile;
  const int tid = (int)threadIdx.x;

  // --- cooperative halo-tile fill with reflect padding ---
  // 1728 elements / 256 threads: 8 iterations for tid < 191, else 7; tail is kept in bounds by
  // clamping the index (duplicate writes of identical data are benign).
  const float* xb = x + (size_t)b * kChan * kH * kW;
  for (int i = tid; i < kChan * kNT + 255; i += 256) {
    const int ii = (i < kChan * kNT) ? i : 0;
    const int c  = ii / kNT;
    const int r  = ii - c * kNT;
    const int ty = r / kHalo;
    const int tx = r - ty * kHalo;
    const int gy = refl(y0 - kPad + ty, kH);
    const int gx = refl(x0 - kPad + tx, kW);
    const float* src = &xb[(c * kH + gy) * kW + gx];
#if defined(USE_ASYNC_LDS)
    // gfx1250 async DMA: global -> LDS, tracked by ASYNCcnt (no VGPR staging)
    __builtin_amdgcn_global_load_async_to_lds_b32(
        (gas1_int_ptr)src, (as3_int_ptr)&tile[ii], 0, 0);
#else
    tile[ii] = *src;
#endif
  }
#if defined(USE_ASYNC_LDS)
# if __has_builtin(__builtin_amdgcn_s_wait_asynccnt)
  __builtin_amdgcn_s_wait_asynccnt(0);
# else
  asm volatile("s_wait_asynccnt 0" ::: "memory");
# endif
#endif
  __syncthreads();

  const int lane = tid & 31;
  const int wave = tid >> 5;
  const int j    = lane & 15;   // WMMA N column (pixel) == A-matrix row (ring)
  const int half = lane >> 4;   // which K-pair this lane supplies

  // Per-lane tap geometry: chunk t covers taps 4t..4t+3 of the 80 non-center
  // taps (center k==40 skipped); this lane supplies taps 4t+2*half and +1.
  // rel* = LDS offset of the tap relative to the pixel's tile base address;
  // Afrag = 0/1 ring-membership indicators (A-matrix), exact in fp32.
  int relA[20], relB[20];
  v2f Afrag[20];
#pragma unroll
  for (int t = 0; t < 20; ++t) {
    const int u0 = 4 * t + 2 * half;
    const int k0 = (u0     < 40) ? u0     : u0 + 1;
    const int k1 = (u0 + 1 < 40) ? u0 + 1 : u0 + 2;
    const int r0 = k0 / 9, q0 = k0 - 9 * r0;
    const int r1 = k1 / 9, q1 = k1 - 9 * r1;
    relA[t] = r0 * kHalo + q0;
    relB[t] = r1 * kHalo + q1;
    const int dy0 = r0 - 4, dx0 = q0 - 4;
    const int dy1 = r1 - 4, dx1 = q1 - 4;
    Afrag[t].x = (c_ridx[dy0 * dy0 + dx0 * dx0] == j) ? 1.0f : 0.0f;
    Afrag[t].y = (c_ridx[dy1 * dy1 + dx1 * dx1] == j) ? 1.0f : 0.0f;
  }

  const float fx = foa[b * 2 + 0];
  const float fy = foa[b * 2 + 1];
  const float inv_diag = rsqrtf((float)(kH * kH + kW * kW));

  for (int rr = 0; rr < 2; ++rr) {
    const int ty = 2 * wave + rr;          // tile row handled by this wave
    const int py = y0 + ty;
    const int px = x0 + j;

    // per-pixel adaptive LoG parameters (both half-lanes compute pixel j)
    const float ddx = (float)px - fx;
    const float ddy = (float)py - fy;
    const float dist = sqrtf(ddx * ddx + ddy * ddy);
    const float dn = dist * inv_diag;
    const float s  = (1.0f - dn) * kSigMin + dn * kSigMax;
    const float s2 = s * s;
    const float a  = 0.5f / s2;                          // tmp = r^2 * a
    const float c0 = -sqrtf(s) * dist / (kPiF * s2 * s2);

    // per-lane gains for its 8 rings (rings 8*half + v), shared by channels
    float g[8];
#pragma unroll
    for (int v = 0; v < 8; ++v) {
      const float q  = c_ringq[8 * half + v];
      const float tq = q * a;
      g[v] = c0 * (1.0f - tq) * expf(-tq);
    }

#if defined(USE_WMMA)
#pragma unroll
    for (int c = 0; c < kChan; ++c) {
      const float* tc = &tile[c * kNT + ty * kHalo + j];
      // Stage 1: batch all B-fragment LDS loads (one dscnt wait, no per-WMMA
      // round trips), then Stage 2: 20 back-to-back chained WMMAs.
      v2f Bfrag[20];
#pragma unroll
      for (int t = 0; t < 20; ++t) {
        Bfrag[t].x = tc[relA[t]];
        Bfrag[t].y = tc[relB[t]];
      }
      v8f acc = {0.f,0.f,0.f,0.f,0.f,0.f,0.f,0.f};
#pragma unroll
      for (int t = 0; t < 20; ++t) {
        // D = A(16 rings x 4 taps) x B(4 taps x 16 pixels) + C, fp32 exact
        acc = __builtin_amdgcn_wmma_f32_16x16x4_f32(
            false, Afrag[t], false, Bfrag[t], (short)0, acc, false, false);
      }
      // D layout: lane j holds rings 0..7, lane j+16 holds rings 8..15 (pixel j)
      float partial = 0.f;
#pragma unroll
      for (int v = 0; v < 8; ++v) partial += g[v] * acc[v];
      partial += __shfl_xor(partial, 16, 32);
      if (half == 0) {
        const float center = tc[4 * kHalo + 4];
        out[(((size_t)b * kChan + c) * kH + py) * kW + px] = partial + c0 * center;
      }
    }
#else
    // VALU fallback: direct per-pixel conv (correct, no WMMA)
    if (half == 0) {
      for (int c = 0; c < kChan; ++c) {
        const float* tc = &tile[c * kNT + ty * kHalo + j];
        float accv = 0.f;
        for (int k = 0; k < 81; ++k) {
          const int r0 = k / 9, q0 = k - 9 * r0;
          const int dy = r0 - 4, dx = q0 - 4;
          const float tq = (float)(dy * dy + dx * dx) * a;
          accv += tc[r0 * kHalo + q0] * (c0 * (1.0f - tq) * expf(-tq));
        }
        out[(((size_t)b * kChan + c) * kH + py) * kW + px] = accv;
      }
    }
#endif
  }
}

extern "C" void kernel_launch(void* const* d_in, const int* in_sizes, int n_in,
                              void* d_out, int out_size, void* d_ws, size_t ws_size,
                              hipStream_t stream) {
  (void)in_sizes; (void)n_in; (void)out_size; (void)d_ws; (void)ws_size;
  const float* x   = (const float*)d_in[0];   // [4,3,256,256] fp32
  const float* foa = (const float*)d_in[1];   // [4,2] fp32
  float* out = (float*)d_out;                 // [4,3,256,256] fp32
  dim3 grid(kW / kTile, kH / kTile, kBatch);  // (16,16,4)
  adaptive_log_conv<<<grid, dim3(256), 0, stream>>>(x, foa, out);
}